// ForwardFunctionalMap_18013092839666
// MI455X (gfx1250) — compile-verified
//
#include <hip/hip_runtime.h>

#define B_    4
#define V_    50000
#define C_    256
#define K_    128
#define LMBDA 100.0f

#define VC_     16                 // V-chunk staged per iteration
#define NCHUNK  (V_ / VC_)         // 3125
#define SPLITV  5                  // V split across blockIdx.z (3125 = 5*625)
#define CPB     (NCHUNK / SPLITV)  // 625 chunks per block (odd: 312 pairs + 1)
#define EPITCH  20                 // evecs LDS row pitch (16+4) -> conflict-free A-frag reads
#define FP2     80                 // feat LDS pair-row pitch in v2f units (2*80 % 64 == 32)

#define BKC ((size_t)B_ * K_ * C_) // 131072 floats
#define BKK ((size_t)B_ * K_ * K_) // 65536 floats

typedef __attribute__((ext_vector_type(2))) float v2f;
typedef __attribute__((ext_vector_type(4))) float v4f;
typedef __attribute__((ext_vector_type(8))) float v8f;
typedef __attribute__((ext_vector_type(4))) int   v4i;

typedef __attribute__((address_space(1))) v4i g_v4i;   // global int4
typedef __attribute__((address_space(3))) v4i l_v4i;   // LDS int4
typedef __attribute__((address_space(1))) int g_i32;   // global int
typedef __attribute__((address_space(3))) int l_i32;   // LDS int

// ---------------------------------------------------------------------------
// Async global->LDS helpers (GLOBAL_LOAD_ASYNC_TO_LDS_*, ASYNCcnt)
// ---------------------------------------------------------------------------
__device__ __forceinline__ void async_g2l_b128(const float* g, float* l) {
#if __has_builtin(__builtin_amdgcn_global_load_async_to_lds_b128)
    __builtin_amdgcn_global_load_async_to_lds_b128((g_v4i*)(void*)g, (l_v4i*)(void*)l, 0, 0);
#else
    *(v4f*)l = *(const v4f*)g;
#endif
}

__device__ __forceinline__ void async_g2l_b32(const float* g, float* l) {
#if __has_builtin(__builtin_amdgcn_global_load_async_to_lds_b32)
    __builtin_amdgcn_global_load_async_to_lds_b32((g_i32*)(void*)g, (l_i32*)(void*)l, 0, 0);
#else
    *l = *g;
#endif
}

template<int N>
__device__ __forceinline__ void wait_async() {
#if __has_builtin(__builtin_amdgcn_s_wait_asynccnt)
    __builtin_amdgcn_s_wait_asynccnt(N);
#elif __has_builtin(__builtin_amdgcn_global_load_async_to_lds_b128)
    asm volatile("s_wait_asynccnt %0" :: "i"(N) : "memory");
#endif
}

// ---------------------------------------------------------------------------
// Kernel 1: partial A = evecs_trans (K x V-slice) @ feat (V-slice x C).
// grid = (B*2, C/64, SPLITV), block = 256 (8 waves). Wave w -> rows
// [16w,16w+16), block.y -> 64-col slab, block.z -> 625-chunk V slice.
// Double-buffered async DMA, software pipeline unrolled x2 so buffer indices
// are compile-time constants. WMMA f32 16x16x4. feat staged K-pair-interleaved
// so each B-frag is one aligned ds_load_b64.
// ---------------------------------------------------------------------------
__global__ __launch_bounds__(256) void fm_gemm_kv(
    const float* __restrict__ feat_x, const float* __restrict__ feat_y,
    const float* __restrict__ evt_x,  const float* __restrict__ evt_y,
    float* __restrict__ P)                    // partials: [z][which][b][K][C]
{
    __shared__ __align__(16) float sE[2][K_ * EPITCH];        // 2 x 10240 B
    __shared__ __align__(16) float sF[2][(VC_/2) * FP2 * 2];  // 2 x  5120 B

    const int bt    = blockIdx.x;
    const int b     = bt >> 1;
    const int which = bt & 1;
    const float* feat = which ? feat_y : feat_x;
    const float* evt  = which ? evt_y  : evt_x;

    const int tid  = threadIdx.x;
    const int wave = tid >> 5;
    const int lane = tid & 31;
    const int k0   = wave * 16;
    const int c0   = blockIdx.y * 64;
    const int hi   = lane >> 4;         // 0 | 1  (koff = 2*hi)
    const int cb   = lane & 15;

    const float* evtB  = evt  + (size_t)b * K_ * V_;
    const float* featB = feat + (size_t)b * V_ * C_;

    v8f acc0 = {}, acc1 = {}, acc2 = {}, acc3 = {};

    // stage one 16-wide V chunk: 2 async b128 (evecs) + 4 async b32 (feat) per thread
    auto issue_chunk = [&](int buf, int chunk) {
        const int v0 = chunk * VC_;
        // evecs tile: 128 rows x 16 floats; q in [0,512): row q/4, 4-float slab q%4
#pragma unroll
        for (int o = 0; o < 2; ++o) {
            const int q = o * 256 + tid;
            const int r = q >> 2;
            const int x = (q & 3) << 2;
            async_g2l_b128(evtB + (size_t)r * V_ + (v0 + x), &sE[buf][r * EPITCH + x]);
        }
        // feat tile: 16 rows x 64 cols, K-pair interleaved:
        // dst[(row/2)*2*FP2 + col*2 + (row&1)] = feat[v0+row][c0+col]
#pragma unroll
        for (int o = 0; o < 4; ++o) {
            const int q   = o * 256 + tid;
            const int row = q >> 6;
            const int col = q & 63;
            async_g2l_b32(featB + (size_t)(v0 + row) * C_ + (c0 + col),
                          &sF[buf][(row >> 1) * (2 * FP2) + col * 2 + (row & 1)]);
        }
    };

    // consume one staged chunk: 4 k-steps x 4 WMMAs
    auto compute_chunk = [&](const float* sEc, const float* sFc) {
#pragma unroll
        for (int vk = 0; vk < VC_; vk += 4) {
            v2f a = *(const v2f*)&sEc[(k0 + cb) * EPITCH + vk + 2 * hi];
            const float* fp = &sFc[((vk >> 1) + hi) * (2 * FP2)];
            v2f b0 = *(const v2f*)&fp[(cb +  0) * 2];
            v2f b1 = *(const v2f*)&fp[(cb + 16) * 2];
            v2f b2 = *(const v2f*)&fp[(cb + 32) * 2];
            v2f b3 = *(const v2f*)&fp[(cb + 48) * 2];
            acc0 = __builtin_amdgcn_wmma_f32_16x16x4_f32(false, a, false, b0, (short)0, acc0, false, false);
            acc1 = __builtin_amdgcn_wmma_f32_16x16x4_f32(false, a, false, b1, (short)0, acc1, false, false);
            acc2 = __builtin_amdgcn_wmma_f32_16x16x4_f32(false, a, false, b2, (short)0, acc2, false, false);
            acc3 = __builtin_amdgcn_wmma_f32_16x16x4_f32(false, a, false, b3, (short)0, acc3, false, false);
        }
    };

    const int cfirst = blockIdx.z * CPB;

    issue_chunk(0, cfirst);
    int c = cfirst;
    for (int it = 0; it < (CPB - 1) / 2; ++it, c += 2) {
        issue_chunk(1, c + 1);
        wait_async<6>();                 // in-order: chunk c (buf0) resident
        __syncthreads();
        compute_chunk(sE[0], sF[0]);
        __syncthreads();

        issue_chunk(0, c + 2);
        wait_async<6>();                 // chunk c+1 (buf1) resident
        __syncthreads();
        compute_chunk(sE[1], sF[1]);
        __syncthreads();
    }
    // final chunk (c == cfirst + CPB - 1, staged in buf0)
    wait_async<0>();
    __syncthreads();
    compute_chunk(sE[0], sF[0]);

    // C/D layout: VGPR r: lanes 0-15 -> M=r, lanes 16-31 -> M=r+8; N = lane&15
    const int mo = hi << 3;
    float* outb = P + (size_t)blockIdx.z * 2 * BKC + (size_t)which * BKC + (size_t)b * K_ * C_;
#pragma unroll
    for (int r = 0; r < 8; ++r) {
        const size_t ro = (size_t)(k0 + r + mo) * C_;
        outb[ro + c0 + cb +  0] = acc0[r];
        outb[ro + c0 + cb + 16] = acc1[r];
        outb[ro + c0 + cb + 32] = acc2[r];
        outb[ro + c0 + cb + 48] = acc3[r];
    }
}

// ---------------------------------------------------------------------------
// Kernel 1b: sum the SPLITV partials -> A (A_x then A_y, contiguous)
// ---------------------------------------------------------------------------
__global__ __launch_bounds__(256) void fm_reduce(
    const float* __restrict__ P, float* __restrict__ A)
{
    const size_t idx = (size_t)blockIdx.x * 256 + threadIdx.x;  // < 2*BKC
    float s = 0.f;
#pragma unroll
    for (int z = 0; z < SPLITV; ++z)
        s += P[(size_t)z * 2 * BKC + idx];
    A[idx] = s;
}

// ---------------------------------------------------------------------------
// Kernel 2: AA_xx = A_x @ A_x^T ; AA_yx = A_y @ A_x^T  (K x K per batch)
// grid = (B*2, K/16, K/16), block = 32 (one wave per 16x16 output tile).
// ---------------------------------------------------------------------------
__global__ __launch_bounds__(32) void fm_gram(
    const float* __restrict__ Ax, const float* __restrict__ Ay,
    float* __restrict__ AAxx, float* __restrict__ AAyx)
{
    const int bt    = blockIdx.x;
    const int b     = bt >> 1;
    const int which = bt & 1;
    const float* L  = which ? Ay   : Ax;   // left matrix (K x C)
    float*       O  = which ? AAyx : AAxx;

    const int lane = threadIdx.x & 31;
    const int k0   = blockIdx.y * 16;
    const int l0   = blockIdx.z * 16;

    const int koff = (lane >> 4) * 2;
    const size_t lbase = (size_t)b * K_ * C_ + (size_t)(k0 + (lane & 15)) * C_;
    const size_t rbase = (size_t)b * K_ * C_ + (size_t)(l0 + (lane & 15)) * C_;

    v8f acc = {};
    for (int c = 0; c < C_; c += 4) {
        v2f a  = *(const v2f*)(L  + lbase + (size_t)(c + koff));
        v2f bf = *(const v2f*)(Ax + rbase + (size_t)(c + koff));
        acc = __builtin_amdgcn_wmma_f32_16x16x4_f32(false, a, false, bf, (short)0, acc, false, false);
    }

    const int mo = (lane >> 4) << 3;
    const int nc = lane & 15;
    float* outb = O + (size_t)b * K_ * K_;
#pragma unroll
    for (int r = 0; r < 8; ++r)
        outb[(size_t)(k0 + r + mo) * K_ + l0 + nc] = acc[r];
}

// ---------------------------------------------------------------------------
// Kernel 3: per (b,m): solve (AA_xx[b] + LMBDA*diag(MASK[b,m,:])) x = AA_yx[b,m,:]
// grid = (K, B), block = 128 (thread i owns row i). Matrix in LDS, pitch 129.
// SPD => Gaussian elimination without pivoting.
// ---------------------------------------------------------------------------
__global__ __launch_bounds__(128) void fm_solve(
    const float* __restrict__ AAxx, const float* __restrict__ AAyx,
    const float* __restrict__ evx,  const float* __restrict__ evy,
    float* __restrict__ out)
{
    extern __shared__ float smem[];
    float* Ml  = smem;                 // K_ * 129
    float* rhs = smem + K_ * 129;      // K_
    float* xs  = rhs + K_;             // K_

    const int m = blockIdx.x;
    const int b = blockIdx.y;
    const int i = threadIdx.x;

    const float* AA = AAxx + (size_t)b * K_ * K_;
    for (int e = i; e < K_ * K_; e += 128) {
        const int r = e >> 7, c = e & 127;
        Ml[r * 129 + c] = AA[e];
    }
    rhs[i] = AAyx[(size_t)b * K_ * K_ + (size_t)m * K_ + i];

    // MASK[b][m][i]; evals sorted ascending -> max is last element
    const float s  = fmaxf(evx[b * K_ + (K_ - 1)], evy[b * K_ + (K_ - 1)]);
    const float g1 = sqrtf(evx[b * K_ + i] / s);   // GAMMA = 0.5
    const float g2 = sqrtf(evy[b * K_ + m] / s);
    const float d1 = g1 * g1 + 1.0f, d2 = g2 * g2 + 1.0f;
    const float re = g2 / d2 - g1 / d1;
    const float im = 1.0f / d2 - 1.0f / d1;
    const float mask = re * re + im * im;

    __syncthreads();
    Ml[i * 129 + i] += LMBDA * mask;
    __syncthreads();

    for (int j = 0; j < K_ - 1; ++j) {
        const float pj = Ml[j * 129 + j];
        const float rj = rhs[j];
        if (i > j) {
            const float f = Ml[i * 129 + j] / pj;
            for (int k = j + 1; k < K_; ++k)
                Ml[i * 129 + k] -= f * Ml[j * 129 + k];
            rhs[i] -= f * rj;
        }
        __syncthreads();
    }

    for (int j = K_ - 1; j >= 0; --j) {
        if (i == j) {
            float a = rhs[j];
            for (int k = j + 1; k < K_; ++k)
                a -= Ml[j * 129 + k] * xs[k];
            xs[j] = a / Ml[j * 129 + j];
        }
        __syncthreads();
    }

    out[(size_t)b * K_ * K_ + (size_t)m * K_ + i] = xs[i];
}

// ---------------------------------------------------------------------------
extern "C" void kernel_launch(void* const* d_in, const int* in_sizes, int n_in,
                              void* d_out, int out_size, void* d_ws, size_t ws_size,
                              hipStream_t stream) {
    (void)in_sizes; (void)n_in; (void)out_size; (void)ws_size;

    const float* feat_x  = (const float*)d_in[0];
    const float* feat_y  = (const float*)d_in[1];
    const float* evals_x = (const float*)d_in[2];
    const float* evals_y = (const float*)d_in[3];
    const float* evt_x   = (const float*)d_in[4];
    const float* evt_y   = (const float*)d_in[5];

    float* ws = (float*)d_ws;
    float* P     = ws;                          // SPLITV * 2 * BKC partials
    float* Afin  = ws + (size_t)SPLITV * 2 * BKC;
    float* Ax    = Afin;
    float* Ay    = Afin + BKC;
    float* AAxx  = Afin + 2 * BKC;
    float* AAyx  = Afin + 2 * BKC + BKK;

    fm_gemm_kv<<<dim3(B_ * 2, C_ / 64, SPLITV), 256, 0, stream>>>(
        feat_x, feat_y, evt_x, evt_y, P);
    fm_reduce<<<dim3((unsigned)((2 * BKC) / 256)), 256, 0, stream>>>(P, Afin);
    fm_gram  <<<dim3(B_ * 2, K_ / 16, K_ / 16), 32, 0, stream>>>(Ax, Ay, AAxx, AAyx);

    const size_t lds_bytes = (size_t)(K_ * 129 + 2 * K_) * sizeof(float);
    fm_solve <<<dim3(K_, B_), 128, lds_bytes, stream>>>(AAxx, AAyx, evals_x, evals_y, (float*)d_out);
}